// MyPosEvaluation_33758442946605
// MI455X (gfx1250) — compile-verified
//
#include <hip/hip_runtime.h>
#include <hip/hip_bf16.h>

// ---------------------------------------------------------------------------
// Reference: mean(|convdata[:, :, 0:3, :, :] - output[:, :, 0:3, :, :]|)
// over shape (128, 100, 6, 32, 32) fp32.
//
// Memory-bound: ~315 MB streamed -> ~13.5 us floor at 23.3 TB/s HBM.
// Strategy: float4 (global_load_b128) grid-stride streaming, per-lane fp32
// accumulation, then a wave32 cross-lane reduction via V_WMMA_F32_16X16X4_F32
// (A = lane partials, B = ones), block partials to scratch, deterministic
// fixed-order final reduce (no fp32 atomics -> bitwise deterministic).
// ---------------------------------------------------------------------------

typedef __attribute__((ext_vector_type(2))) float v2f;
typedef __attribute__((ext_vector_type(8))) float v8f;

#define NBLOCKS   1024
#define NTHREADS  256
#define NWAVES    (NTHREADS / 32)

#define REGION_ELEMS 39321600u            // 128*100*3*32*32
#define REGION_V4    (REGION_ELEMS / 4u)  // 9,830,400 float4s in the used region
#define CHUNK_V4     768u                 // 3072 used floats per (b,t) pair / 4
#define PERIOD_V4    1536u                // 6144 floats per (b,t) pair / 4
#define INV_N        (1.0f / 39321600.0f)

// Wave32 sum of one float per lane using WMMA f32 16x16x4:
//   A[16x4]: lane L (L<16) holds {A[L][0]=x, A[L][1]=0};
//            lane L (>=16) holds {A[L-16][2]=x, A[L-16][3]=0}.
//   B[4x16] = all ones  (layout-independent since every element is 1.0).
//   D[m][n] = partial[m] + partial[m+16]  for every column n.
// Lane 0 holds D[0..7][0] in its 8 C VGPRs; lane 16 holds D[8..15][0].
// lane0_sum + lane16_sum = sum over all 32 lanes. Combine with one shuffle.
// Requires EXEC == all ones (callers are in converged, full-wave control flow).
__device__ __forceinline__ float wave_sum32_wmma(float x) {
  v2f a;
  a[0] = x;
  a[1] = 0.0f;
  v2f b;
  b[0] = 1.0f;
  b[1] = 1.0f;
  v8f c = {};
  c = __builtin_amdgcn_wmma_f32_16x16x4_f32(
      /*neg_a=*/false, a, /*neg_b=*/false, b,
      /*c_mod=*/(short)0, c, /*reuse_a=*/false, /*reuse_b=*/false);
  float s = c[0] + c[1] + c[2] + c[3] + c[4] + c[5] + c[6] + c[7];
  float s_hi = __shfl(s, 16, 32);   // lane 16's half-sum, broadcast to all
  return s + s_hi;                  // valid total in lane 0
}

__global__ void absdiff_partial_kernel(const float* __restrict__ outp,
                                       const float* __restrict__ conv,
                                       float* __restrict__ partials) {
  __shared__ float lds[NWAVES];

  const unsigned tid    = threadIdx.x;
  const unsigned gtid   = blockIdx.x * NTHREADS + tid;
  const unsigned stride = NBLOCKS * NTHREADS;

  const float4* __restrict__ o4 = (const float4*)outp;
  const float4* __restrict__ c4 = (const float4*)conv;

  float acc = 0.0f;
  // Used region = chunks of 3072 floats (channels 0:3) every 6144 floats.
  for (unsigned v = gtid; v < REGION_V4; v += stride) {
    unsigned chunk = v / CHUNK_V4;
    unsigned off   = v - chunk * CHUNK_V4;
    unsigned idx   = chunk * PERIOD_V4 + off;   // float4 index, < 2^31
    float4 a = o4[idx];
    float4 b = c4[idx];
    acc += fabsf(b.x - a.x) + fabsf(b.y - a.y) +
           fabsf(b.z - a.z) + fabsf(b.w - a.w);
  }

  // Converged here: EXEC all ones -> WMMA reduce across the wave.
  float wtot = wave_sum32_wmma(acc);

  const unsigned lane = tid & 31u;
  const unsigned wave = tid >> 5u;
  if (lane == 0) lds[wave] = wtot;
  __syncthreads();

  if (tid == 0) {
    float bsum = 0.0f;
#pragma unroll
    for (int i = 0; i < NWAVES; ++i) bsum += lds[i];  // fixed order: deterministic
    partials[blockIdx.x] = bsum;
  }
}

// One full wave32 (EXEC all ones). Each lane sums a fixed strided subset of
// the block partials (deterministic order), then WMMA-reduces across lanes.
__global__ void finalize_kernel(const float* __restrict__ partials,
                                float* __restrict__ out) {
  const unsigned lane = threadIdx.x;  // 0..31
  float acc = 0.0f;
  for (unsigned i = lane; i < NBLOCKS; i += 32u) acc += partials[i];
  float tot = wave_sum32_wmma(acc);
  if (lane == 0) out[0] = tot * INV_N;
}

extern "C" void kernel_launch(void* const* d_in, const int* in_sizes, int n_in,
                              void* d_out, int out_size, void* d_ws, size_t ws_size,
                              hipStream_t stream) {
  const float* outp = (const float*)d_in[0];   // "output"
  const float* conv = (const float*)d_in[1];   // "convdata"
  float* partials   = (float*)d_ws;            // NBLOCKS * 4 bytes = 4 KB
  float* result     = (float*)d_out;

  absdiff_partial_kernel<<<NBLOCKS, NTHREADS, 0, stream>>>(outp, conv, partials);
  finalize_kernel<<<1, 32, 0, stream>>>(partials, result);
}